// SlicedVectorQuantize_3272765079614
// MI455X (gfx1250) — compile-verified
//
#include <hip/hip_runtime.h>
#include <math.h>

// Problem constants (match reference)
#define BATCH 16
#define TDIM  4096
#define DDIM  512
#define SUBD  256
#define KC    1024
#define NVEC  (BATCH * TDIM)                 // 65536 vectors
#define NOUT  ((size_t)BATCH * DDIM * TDIM)  // 33,554,432 quant elements

typedef __attribute__((ext_vector_type(2))) float v2f;
typedef __attribute__((ext_vector_type(8))) float v8f;

#define THREADS_VQ 64     // 2 waves (wave32); each wave owns 16 rows
#define ROWS_PER_BLOCK 32
#define ESTRIDE 260       // 256 + 4 pad: rows stay 16B-aligned, banks 4*col+kb distinct

// gfx1250 async global->LDS path (ASYNCcnt); fall back to plain copy if absent.
#if __has_builtin(__builtin_amdgcn_global_load_async_to_lds_b128) && \
    __has_builtin(__builtin_amdgcn_s_wait_asynccnt)
#define USE_ASYNC_LDS 1
// Builtin prototype (from clang diagnostic): (int4 AS1* src, int4 AS3* dst, Ii, Ii)
typedef int gv4i __attribute__((vector_size(16)));
typedef __attribute__((address_space(1))) gv4i* gv4i_g;
typedef __attribute__((address_space(3))) gv4i* gv4i_l;
#else
#define USE_ASYNC_LDS 0
#endif

// ---------------------------------------------------------------------------
// Kernel 0: zero ws accumulators (counts + sum-of-squares); ws is poisoned.
// ---------------------------------------------------------------------------
__global__ void vq_zero_ws(float* cnt1, float* cnt2, float* sumsq) {
    int i = blockIdx.x * blockDim.x + threadIdx.x;
    if (i < KC)            cnt1[i] = 0.0f;
    else if (i < 2 * KC)   cnt2[i - KC] = 0.0f;
    else if (i == 2 * KC)  *sumsq = 0.0f;
}

// ---------------------------------------------------------------------------
// Kernel 1: ce[k] = 0.5 * ||e_k||^2  (argmin dist == argmax (f.e - ce))
// ---------------------------------------------------------------------------
__global__ void vq_code_norms(const float* __restrict__ emb1,
                              const float* __restrict__ emb2,
                              float* __restrict__ ce1,
                              float* __restrict__ ce2) {
    int k = blockIdx.x * blockDim.x + threadIdx.x;   // 0 .. 2047
    const float* e = (k < KC) ? (emb1 + (size_t)k * SUBD)
                              : (emb2 + (size_t)(k - KC) * SUBD);
    float s = 0.0f;
    #pragma unroll 8
    for (int d = 0; d < SUBD; ++d) { float v = e[d]; s += v * v; }
    if (k < KC) ce1[k] = 0.5f * s;
    else        ce2[k - KC] = 0.5f * s;
}

// ---------------------------------------------------------------------------
// Kernel 2: WMMA nearest-codeword search.
//   grid.x = NVEC/32 row tiles, grid.y = 2 halves.
//   A (16 rows x 256 feats per wave) lives entirely in VGPRs (128/lane).
//   B (16-code x 256-feat tiles) double-buffered in LDS via async copies.
// ---------------------------------------------------------------------------
__global__ __launch_bounds__(THREADS_VQ)
void vq_argmin(const float* __restrict__ x,
               const float* __restrict__ emb1, const float* __restrict__ emb2,
               const float* __restrict__ ce1,  const float* __restrict__ ce2,
               float* cnt1, float* cnt2,
               int* __restrict__ idx1, int* __restrict__ idx2) {
    __shared__ __align__(16) float Es[2][16 * ESTRIDE];
    __shared__ float ceS[KC];

    const int tid  = threadIdx.x;
    const int lane = tid & 31;
    const int wave = tid >> 5;
    const int h    = blockIdx.y;
    const int row0 = blockIdx.x * ROWS_PER_BLOCK;
    const int b    = row0 / TDIM;
    const int t0   = row0 % TDIM;

    const float* xh   = x + (size_t)b * DDIM * TDIM + (size_t)h * SUBD * TDIM + t0;
    const float* eh   = h ? emb2 : emb1;
    const float* ceh  = h ? ce2  : ce1;
    float*       cnth = h ? cnt2 : cnt1;
    int*         idxh = h ? idx2 : idx1;

    // A fragments straight into registers:
    //   lane&15 = row M; lanes<16 hold K={0,1} mod 4, lanes>=16 hold K={2,3}.
    const int   koff = (lane >> 4) * 2;
    const float* xr  = xh + (wave * 16 + (lane & 15));   // + d*TDIM per feature
    float areg[128];
    #pragma unroll
    for (int kk = 0; kk < SUBD / 4; ++kk) {
        areg[2 * kk]     = xr[(size_t)(4 * kk + koff) * TDIM];
        areg[2 * kk + 1] = xr[(size_t)(4 * kk + koff + 1) * TDIM];
    }

    for (int i = tid; i < KC; i += THREADS_VQ) ceS[i] = ceh[i];

#if USE_ASYNC_LDS
    // Stage one 16x256 codebook tile (16 KB) with b128 async chunks.
    auto issue_tile = [&](int ct, int buf) {
        const float* src = eh + (size_t)ct * 16 * SUBD;
        #pragma unroll
        for (int i = 0; i < 16; ++i) {                  // 1024 chunks / 64 threads
            int chunk = i * THREADS_VQ + tid;
            int code  = chunk >> 6;                     // 64 x 16B chunks per row
            int woff  = (chunk & 63) * 4;               // float offset in row
            const float* g = src + code * SUBD + woff;
            float*       l = &Es[buf][code * ESTRIDE + woff];
            __builtin_amdgcn_global_load_async_to_lds_b128(
                (gv4i_g)(unsigned long long)g,
                (gv4i_l)(unsigned int)(unsigned long long)l,
                0, 0);
        }
    };
#endif

    const int col = lane & 15;
    float bestv[8];
    int   besti[8];
    #pragma unroll
    for (int i = 0; i < 8; ++i) { bestv[i] = -3.4e38f; besti[i] = 0; }

#if USE_ASYNC_LDS
    issue_tile(0, 0);
    for (int ct = 0; ct < KC / 16; ++ct) {
        __builtin_amdgcn_s_wait_asynccnt(0);   // this wave's tile-ct chunks landed
        __syncthreads();                       // all waves landed + prev compute done
        if (ct + 1 < KC / 16) issue_tile(ct + 1, (ct + 1) & 1);
        const float* Eb = Es[ct & 1];
#else
    for (int ct = 0; ct < KC / 16; ++ct) {
        __syncthreads();                       // previous tile fully consumed
        if (ct + 1 < KC / 16)
            __builtin_prefetch(eh + (size_t)(ct + 1) * 16 * SUBD, 0, 0);
        #pragma unroll 4
        for (int i = 0; i < (16 * SUBD) / THREADS_VQ; ++i) {
            int lin = i * THREADS_VQ + tid;
            int c   = lin >> 8;
            int d   = lin & (SUBD - 1);
            Es[0][c * ESTRIDE + d] = eh[(size_t)(ct * 16 + c) * SUBD + d];
        }
        __syncthreads();
        const float* Eb = Es[0];
#endif

        v8f acc = {0.f, 0.f, 0.f, 0.f, 0.f, 0.f, 0.f, 0.f};
        #pragma unroll
        for (int kk = 0; kk < SUBD / 4; ++kk) {
            int kb = kk * 4 + koff;
            v2f av, bv;
            av[0] = areg[2 * kk];
            av[1] = areg[2 * kk + 1];
            bv[0] = Eb[col * ESTRIDE + kb];
            bv[1] = Eb[col * ESTRIDE + kb + 1];
            acc = __builtin_amdgcn_wmma_f32_16x16x4_f32(
                false, av, false, bv, (short)0, acc, false, false);
        }

        const int   gcol = ct * 16 + col;
        const float cv   = ceS[gcol];
        #pragma unroll
        for (int i = 0; i < 8; ++i) {
            float s = acc[i] - cv;               // maximize f.e - 0.5||e||^2
            if (s > bestv[i]) { bestv[i] = s; besti[i] = gcol; }
        }
    }

    // Reduce across the 16 lanes of each half-wave (rows live per-half)
    #pragma unroll
    for (int m = 1; m <= 8; m <<= 1) {
        #pragma unroll
        for (int i = 0; i < 8; ++i) {
            float ov = __shfl_xor(bestv[i], m, 32);
            int   oi = __shfl_xor(besti[i], m, 32);
            if (ov > bestv[i] || (ov == bestv[i] && oi < besti[i])) {
                bestv[i] = ov; besti[i] = oi;
            }
        }
    }

    if ((lane & 15) == 0) {
        int half = lane >> 4;                    // 0 -> rows 0..7, 1 -> rows 8..15
        #pragma unroll
        for (int i = 0; i < 8; ++i) {
            int rg = row0 + wave * 16 + half * 8 + i;
            idxh[rg] = besti[i];
            atomicAdd(&cnth[besti[i]], 1.0f);
        }
    }
}

// ---------------------------------------------------------------------------
// Kernel 3: gather codewords, write straight-through output [B,D,T] (float4
//           vectorized along t), accumulate sum of (quant - x)^2.
// ---------------------------------------------------------------------------
__global__ __launch_bounds__(256)
void vq_quant_loss(const float* __restrict__ x,
                   const float* __restrict__ emb1, const float* __restrict__ emb2,
                   const int* __restrict__ idx1,  const int* __restrict__ idx2,
                   float* __restrict__ out, float* sumsq) {
    size_t vec = (size_t)blockIdx.x * 256 + threadIdx.x;   // over B*D*T/4
    size_t lin = vec * 4;
    int t = (int)(lin % TDIM);
    int d = (int)((lin / TDIM) % DDIM);
    int b = (int)(lin / ((size_t)TDIM * DDIM));
    int n = b * TDIM + t;

    float4 xv = *(const float4*)(x + lin);
    float q[4];
    #pragma unroll
    for (int j = 0; j < 4; ++j) {
        if (d < SUBD) q[j] = emb1[(size_t)idx1[n + j] * SUBD + d];
        else          q[j] = emb2[(size_t)idx2[n + j] * SUBD + (d - SUBD)];
    }

    float4 o;
    o.x = xv.x + (q[0] - xv.x);                  // straight-through forward value
    o.y = xv.y + (q[1] - xv.y);
    o.z = xv.z + (q[2] - xv.z);
    o.w = xv.w + (q[3] - xv.w);
    *(float4*)(out + lin) = o;

    float d0 = q[0] - xv.x, d1 = q[1] - xv.y, d2 = q[2] - xv.z, d3 = q[3] - xv.w;
    float v = d0 * d0 + d1 * d1 + d2 * d2 + d3 * d3;

    #pragma unroll
    for (int m = 16; m >= 1; m >>= 1) v += __shfl_xor(v, m, 32);
    __shared__ float red[8];
    int lane = threadIdx.x & 31, w = threadIdx.x >> 5;
    if (lane == 0) red[w] = v;
    __syncthreads();
    if (threadIdx.x == 0) {
        float s = 0.f;
        #pragma unroll
        for (int i = 0; i < 8; ++i) s += red[i];
        atomicAdd(sumsq, s);
    }
}

// ---------------------------------------------------------------------------
// Kernel 4: finalize loss + perplexity scalars
// ---------------------------------------------------------------------------
__global__ __launch_bounds__(1024)
void vq_finalize(const float* __restrict__ cnt1, const float* __restrict__ cnt2,
                 const float* __restrict__ sumsq, float* out_scalars) {
    __shared__ float s1s[32], s2s[32];
    int tid = threadIdx.x, lane = tid & 31, w = tid >> 5;
    float p1 = cnt1[tid] / (float)NVEC;
    float p2 = cnt2[tid] / (float)NVEC;
    float e1 = p1 * logf(p1 + 1e-10f);
    float e2 = p2 * logf(p2 + 1e-10f);
    #pragma unroll
    for (int m = 16; m >= 1; m >>= 1) {
        e1 += __shfl_xor(e1, m, 32);
        e2 += __shfl_xor(e2, m, 32);
    }
    if (lane == 0) { s1s[w] = e1; s2s[w] = e2; }
    __syncthreads();
    if (tid == 0) {
        float a = 0.f, c = 0.f;
        #pragma unroll
        for (int i = 0; i < 32; ++i) { a += s1s[i]; c += s2s[i]; }
        out_scalars[0] = 1.25f * (*sumsq) / (float)((size_t)NVEC * DDIM);
        out_scalars[1] = expf(-a) + expf(-c);
    }
}

// ---------------------------------------------------------------------------
// Launch
// ---------------------------------------------------------------------------
extern "C" void kernel_launch(void* const* d_in, const int* in_sizes, int n_in,
                              void* d_out, int out_size, void* d_ws, size_t ws_size,
                              hipStream_t stream) {
    const float* x    = (const float*)d_in[0];
    const float* emb1 = (const float*)d_in[1];
    const float* emb2 = (const float*)d_in[2];
    float* out = (float*)d_out;

    // ws layout (floats): ce1[1024] ce2[1024] cnt1[1024] cnt2[1024] sumsq[1]
    //                     pad, idx1[65536] idx2[65536]  (~530 KB total)
    float* ws    = (float*)d_ws;
    float* ce1   = ws;
    float* ce2   = ws + 1024;
    float* cnt1  = ws + 2048;
    float* cnt2  = ws + 3072;
    float* sumsq = ws + 4096;
    int*   idx1  = (int*)(ws + 4352);
    int*   idx2  = idx1 + NVEC;

    vq_zero_ws<<<(2 * KC + 1 + 255) / 256, 256, 0, stream>>>(cnt1, cnt2, sumsq);
    vq_code_norms<<<(2 * KC) / 256, 256, 0, stream>>>(emb1, emb2, ce1, ce2);

    dim3 grid(NVEC / ROWS_PER_BLOCK, 2);
    vq_argmin<<<grid, THREADS_VQ, 0, stream>>>(x, emb1, emb2, ce1, ce2,
                                               cnt1, cnt2, idx1, idx2);

    vq_quant_loss<<<(unsigned)(NOUT / 4 / 256), 256, 0, stream>>>(
        x, emb1, emb2, idx1, idx2, out, sumsq);

    vq_finalize<<<1, 1024, 0, stream>>>(cnt1, cnt2, sumsq, out + NOUT);
}